// InterAttention_33775622816444
// MI455X (gfx1250) — compile-verified
//
#include <hip/hip_runtime.h>

// ---------------------------------------------------------------------------
// InterAttention, MI455X (gfx1250, wave32, WMMA f32 16x16x4).
// Memory-bound (~600MB min traffic => ~26us @ 23.3TB/s vs ~52 GFLOP), so we
// keep exact fp32 via V_WMMA_F32_16X16X4_F32 instead of casting to bf16.
// Workspace layout (floats):
//   v_ws  [8*192][256][64]  = 25,165,824   (V in [b][c][win][tok])
//   o_ws  [8*192][256][64]  = 25,165,824   (attn output, same layout)
//   qg_ws [8*192][256]      =    393,216
//   kg_ws [8*192][256]      =    393,216
// total ~204.5 MB of d_ws.
// ---------------------------------------------------------------------------

#define B8    8
#define NWIN  256
#define NTOK  64      // 8x8 tokens per window
#define DIM   192
#define QKVC  576

typedef float v2f __attribute__((ext_vector_type(2)));
typedef float v8f __attribute__((ext_vector_type(8)));

__device__ __forceinline__ v8f wmma_f32(v2f a, v2f b, v8f c) {
  // D = A(16x4) * B(4x16) + C(16x16), exact fp32
  return __builtin_amdgcn_wmma_f32_16x16x4_f32(false, a, false, b,
                                               (short)0, c, false, false);
}

// ---------------------------------------------------------------------------
// Kernel A: fused QKV GEMM + bias + q-scale + gradient-magnitude reduction.
// One block per window. q,k never leave the WGP: only their grad scalars and
// the V slice are written out.
// ---------------------------------------------------------------------------
__global__ __launch_bounds__(128) void qkv_grad_kernel(
    const float* __restrict__ x, const float* __restrict__ qkv_w,
    const float* __restrict__ qkv_b, float* __restrict__ v_ws,
    float* __restrict__ qg_ws, float* __restrict__ kg_ws) {
  __shared__ float xs[NTOK * 193];   // A tile [t][k], padded row stride
  __shared__ float wbuf[16 * 193];   // weight chunk [cout][k]; reused as
                                     // [t][cout] (stride 17) epilogue staging

  const int w    = blockIdx.x;       // b*NWIN + i
  const int b    = w >> 8;
  const int i    = w & 255;
  const int tid  = threadIdx.x;
  const int lane = tid & 31;
  const int wave = tid >> 5;         // 0..3 -> M tile (16 tokens)
  const int half = lane >> 4;        // K sub-pair select per ISA layout
  const int ln   = lane & 15;

  const float* xrow = x + (size_t)w * NTOK * DIM;
  for (int e = tid; e < NTOK * DIM; e += 128) {
    int t = e / DIM, k = e - t * DIM;
    xs[t * 193 + k] = xrow[e];
  }

  const float qscale = 0.17677669529663687f;  // 32^-0.5

  for (int chunk = 0; chunk < QKVC / 16; ++chunk) {
    const int cout0 = chunk * 16;
    for (int e = tid; e < 16 * DIM; e += 128) {
      int cl = e / DIM, k = e - cl * DIM;
      wbuf[cl * 193 + k] = qkv_w[(size_t)(cout0 + cl) * DIM + k];
    }
    __syncthreads();

    v8f acc = {};
    const float* arow = &xs[(wave * 16 + ln) * 193];
    const float* brow = &wbuf[ln * 193];
    for (int k0 = 0; k0 < DIM; k0 += 4) {
      int kk = k0 + half * 2;
      v2f a  = {arow[kk], arow[kk + 1]};
      v2f bb = {brow[kk], brow[kk + 1]};
      acc = wmma_f32(a, bb, acc);
    }
    __syncthreads();  // all WMMAs done -> wbuf reusable as staging

    const bool  isQ  = (cout0 < DIM);
    const bool  isV  = (cout0 >= 2 * DIM);
    const float bias = qkv_b[cout0 + ln];
    const float sc   = isQ ? qscale : 1.0f;
#pragma unroll
    for (int g = 0; g < 8; ++g) {
      int t = wave * 16 + half * 8 + g;         // D layout: M = g (+8 hi half)
      wbuf[t * 17 + ln] = (acc[g] + bias) * sc;
    }
    __syncthreads();

    if (!isV) {
      // |gx|+|gy| over the 8x8 grid, zero pre-pad (matches reference)
      if (tid < 16) {
        float s = 0.0f;
        for (int y = 0; y < 8; ++y)
          for (int xx = 0; xx < 8; ++xx) {
            int t = y * 8 + xx;
            float cur  = wbuf[t * 17 + tid];
            float left = xx ? wbuf[(t - 1) * 17 + tid] : 0.0f;
            float up   = y ? wbuf[(t - 8) * 17 + tid] : 0.0f;
            s += fabsf(cur - left) + fabsf(cur - up);
          }
        int c = isQ ? (cout0 + tid) : (cout0 - DIM + tid);
        float* dst = isQ ? qg_ws : kg_ws;
        dst[((size_t)b * DIM + c) * NWIN + i] = s;
      }
    } else {
      // V chunk -> [b][c][win][tok]: contiguous 64-float runs per channel
      for (int e = tid; e < 16 * NTOK; e += 128) {
        int cl = e >> 6, t = e & 63;
        int c = cout0 - 2 * DIM + cl;
        v_ws[(((size_t)b * DIM + c) * NWIN + i) * NTOK + t] = wbuf[t * 17 + cl];
      }
    }
    __syncthreads();
  }
}

// ---------------------------------------------------------------------------
// Kernel B: fused rank-1-score softmax + (attn @ V) per (b, head, d) group.
// Two-pass softmax: m_i = a_i>=0 ? a_i*max(k) : a_i*min(k); Z_i exact;
// then 32-wide K blocks: exp tile + V tile in LDS, WMMA accumulate,
// row-scale by 1/Z at the end. The 400MB attn tensor is never materialized.
// ---------------------------------------------------------------------------
__global__ __launch_bounds__(256) void attn_kernel(
    const float* __restrict__ v_ws, const float* __restrict__ qg_ws,
    const float* __restrict__ kg_ws, float* __restrict__ o_ws) {
  __shared__ float kgs[NWIN];
  __shared__ float ags[NWIN];
  __shared__ float ms[NWIN];
  __shared__ float zs[NWIN];          // 1/Z
  __shared__ float Es[NWIN * 33];     // exp block [i][jl], padded
  __shared__ float Vs[32 * NTOK];     // V block [jl][t]

  const int bc   = blockIdx.x;        // b*DIM + (h*32+d)
  const int tid  = threadIdx.x;
  const int lane = tid & 31;
  const int wave = tid >> 5;          // 0..7 -> two M tiles each
  const int half = lane >> 4;
  const int ln   = lane & 15;

  kgs[tid] = kg_ws[(size_t)bc * NWIN + tid];
  ags[tid] = qg_ws[(size_t)bc * NWIN + tid];
  __syncthreads();

  float kmax = -3.402823466e38f, kmin = 3.402823466e38f;
  for (int j = 0; j < NWIN; ++j) {
    float kv = kgs[j];
    kmax = fmaxf(kmax, kv);
    kmin = fminf(kmin, kv);
  }
  {
    float a = ags[tid];
    float m = (a >= 0.0f) ? a * kmax : a * kmin;
    float z = 0.0f;
    for (int j = 0; j < NWIN; ++j) z += __expf(a * kgs[j] - m);
    ms[tid] = m;
    zs[tid] = 1.0f / z;
  }
  __syncthreads();

  v8f acc[8];
#pragma unroll
  for (int q = 0; q < 8; ++q) acc[q] = (v8f){};

  const float a_row = ags[tid];
  const float m_row = ms[tid];
  const float* vbase = v_ws + (size_t)bc * NWIN * NTOK;

  for (int jb = 0; jb < NWIN; jb += 32) {
    // thread tid owns softmax row i = tid
#pragma unroll 4
    for (int jl = 0; jl < 32; ++jl)
      Es[tid * 33 + jl] = __expf(a_row * kgs[jb + jl] - m_row);
    for (int e = tid; e < 32 * NTOK; e += 256)  // contiguous global load
      Vs[e] = vbase[jb * NTOK + e];
    __syncthreads();

    for (int k0 = 0; k0 < 32; k0 += 4) {
      int kk = k0 + half * 2;
      v2f bm[4];
#pragma unroll
      for (int nt = 0; nt < 4; ++nt) {
        int col = nt * 16 + ln;
        bm[nt].x = Vs[kk * NTOK + col];
        bm[nt].y = Vs[(kk + 1) * NTOK + col];
      }
#pragma unroll
      for (int mtl = 0; mtl < 2; ++mtl) {
        int row = (wave * 2 + mtl) * 16 + ln;
        v2f a = {Es[row * 33 + kk], Es[row * 33 + kk + 1]};
#pragma unroll
        for (int nt = 0; nt < 4; ++nt)
          acc[mtl * 4 + nt] = wmma_f32(a, bm[nt], acc[mtl * 4 + nt]);
      }
    }
    __syncthreads();
  }

#pragma unroll
  for (int mtl = 0; mtl < 2; ++mtl)
#pragma unroll
    for (int nt = 0; nt < 4; ++nt)
#pragma unroll
      for (int g = 0; g < 8; ++g) {
        int irow = (wave * 2 + mtl) * 16 + half * 8 + g;
        int t    = nt * 16 + ln;
        o_ws[((size_t)bc * NWIN + irow) * NTOK + t] =
            acc[mtl * 4 + nt][g] * zs[irow];
      }
}

// ---------------------------------------------------------------------------
// Kernel C: output projection. One block per window; gathers the attention
// output tile with 256B-contiguous runs into channel-major LDS (natural
// A-operand layout), WMMA proj GEMM, fused bias, coalesced store.
// ---------------------------------------------------------------------------
__global__ __launch_bounds__(128) void proj_kernel(
    const float* __restrict__ o_ws, const float* __restrict__ proj_w,
    const float* __restrict__ proj_b, float* __restrict__ out) {
  __shared__ float ots[DIM * NTOK];   // [c][t]
  __shared__ float wts[16 * 193];

  const int w    = blockIdx.x;
  const int b    = w >> 8;
  const int i    = w & 255;
  const int tid  = threadIdx.x;
  const int lane = tid & 31;
  const int wave = tid >> 5;
  const int half = lane >> 4;
  const int ln   = lane & 15;

  for (int e = tid; e < DIM * NTOK; e += 128) {
    int c = e >> 6, t = e & 63;
    ots[e] = o_ws[(((size_t)b * DIM + c) * NWIN + i) * NTOK + t];
  }

  for (int chunk = 0; chunk < DIM / 16; ++chunk) {
    const int cout0 = chunk * 16;
    for (int e = tid; e < 16 * DIM; e += 128) {
      int cl = e / DIM, k = e - cl * DIM;
      wts[cl * 193 + k] = proj_w[(size_t)(cout0 + cl) * DIM + k];
    }
    __syncthreads();

    v8f acc = {};
    for (int k0 = 0; k0 < DIM; k0 += 4) {
      int kk = k0 + half * 2;
      v2f a  = {ots[kk * NTOK + wave * 16 + ln],
                ots[(kk + 1) * NTOK + wave * 16 + ln]};
      v2f bb = {wts[ln * 193 + kk], wts[ln * 193 + kk + 1]};
      acc = wmma_f32(a, bb, acc);
    }

    const float bias = proj_b[cout0 + ln];
#pragma unroll
    for (int g = 0; g < 8; ++g) {
      int t = wave * 16 + half * 8 + g;
      out[((size_t)w * NTOK + t) * DIM + cout0 + ln] = acc[g] + bias;
    }
    __syncthreads();
  }
}

// ---------------------------------------------------------------------------
extern "C" void kernel_launch(void* const* d_in, const int* in_sizes, int n_in,
                              void* d_out, int out_size, void* d_ws,
                              size_t ws_size, hipStream_t stream) {
  const float* x      = (const float*)d_in[0];
  const float* qkv_w  = (const float*)d_in[1];
  const float* qkv_b  = (const float*)d_in[2];
  const float* proj_w = (const float*)d_in[3];
  const float* proj_b = (const float*)d_in[4];
  float* out = (float*)d_out;

  float* ws    = (float*)d_ws;
  const size_t VO = (size_t)B8 * DIM * NWIN * NTOK;  // 25,165,824 floats
  float* v_ws  = ws;
  float* o_ws  = ws + VO;
  float* qg_ws = ws + 2 * VO;
  float* kg_ws = qg_ws + (size_t)B8 * DIM * NWIN;

  qkv_grad_kernel<<<B8 * NWIN, 128, 0, stream>>>(x, qkv_w, qkv_b, v_ws, qg_ws,
                                                 kg_ws);
  attn_kernel<<<B8 * DIM, 256, 0, stream>>>(v_ws, qg_ws, kg_ws, o_ws);
  proj_kernel<<<B8 * NWIN, 128, 0, stream>>>(o_ws, proj_w, proj_b, out);
}